// HomoGNNLayer_790273982770
// MI455X (gfx1250) — compile-verified
//
#include <hip/hip_runtime.h>

// ---------------------------------------------------------------------------
// SAGEConv(project=True, aggr=sum) + LayerNorm for MI455X (gfx1250, wave32).
// fp32 WMMA (V_WMMA_F32_16X16X4_F32) for all GEMMs: exact-precision matrix
// pipe. Edge scatter via global_atomic_add_f32 — agg/h are 51 MB each and
// stay resident in the 192 MB L2, so the scatter runs at L2 atomic rate.
// ---------------------------------------------------------------------------

typedef __attribute__((ext_vector_type(2))) float v2f;
typedef __attribute__((ext_vector_type(8))) float v8f;

#define C_DIM 128
#define LN_EPS 1e-5f
#define LDS_STRIDE 132  // 128 + 4 pad: rows 8 apart land 32 banks apart

static __device__ __forceinline__ v8f wmma_f32_k4(v2f a, v2f b, v8f c) {
  // D = A(16x4,f32) * B(4x16,f32) + C(16x16,f32)
  return __builtin_amdgcn_wmma_f32_16x16x4_f32(
      /*neg_a=*/false, a, /*neg_b=*/false, b,
      /*c_mod=*/(short)0, c, /*reuse_a=*/false, /*reuse_b=*/false);
}

// ---------------------------------------------------------------------------
// K1: h = relu(x @ W_proj^T + b_proj)
// Block: 16 node-rows, 8 waves -> 8 column tiles of 16. K-loop: 32 x wmma.
// A frag (16x4 f32): lanes 0-15 rows M=lane K={k0,k0+1}; lanes 16-31 same M,
// K={k0+2,k0+3}  -> per-lane float2 load. B frag mirrors A (row-striped).
// ---------------------------------------------------------------------------
__global__ __launch_bounds__(256) void sage_proj_relu(
    const float* __restrict__ x, const float* __restrict__ Wp,
    const float* __restrict__ bp, float* __restrict__ h, int N) {
  const int wave  = threadIdx.x >> 5;
  const int lane  = threadIdx.x & 31;
  const int l16   = lane & 15;
  const int khalf = (lane >> 4) * 2;       // 0 or 2
  const int row0  = blockIdx.x * 16;       // N % 16 == 0 for this problem
  const int col   = wave * 16 + l16;

  const float* arow = x  + (size_t)(row0 + l16) * C_DIM;
  const float* brow = Wp + (size_t)col * C_DIM;   // B[k][col] = Wp[col*C + k]

  v8f acc = {};
#pragma unroll
  for (int k0 = 0; k0 < C_DIM; k0 += 4) {
    v2f a = *(const v2f*)(arow + k0 + khalf);
    v2f b = *(const v2f*)(brow + k0 + khalf);
    acc = wmma_f32_k4(a, b, acc);
  }

  const float bias = bp[col];
  const int rofs = (lane >> 4) * 8;        // C/D layout: lanes 16-31 -> M+8
  float* out = h + (size_t)row0 * C_DIM + col;
#pragma unroll
  for (int i = 0; i < 8; ++i) {
    float v = acc[i] + bias;
    out[(size_t)(i + rofs) * C_DIM] = v > 0.f ? v : 0.f;
  }
}

// ---------------------------------------------------------------------------
// K2: zero the aggregation buffer (workspace is poisoned by the harness).
// ---------------------------------------------------------------------------
__global__ __launch_bounds__(256) void sage_zero(float4* __restrict__ p,
                                                 long n4) {
  long i = (long)blockIdx.x * blockDim.x + threadIdx.x;
  long stride = (long)gridDim.x * blockDim.x;
  float4 z = {0.f, 0.f, 0.f, 0.f};
  for (; i < n4; i += stride) p[i] = z;
}

// ---------------------------------------------------------------------------
// K3: edge scatter-sum. One edge per wave; lane handles 4 channels.
// float4 gather from h[src] (L2-resident), 4 fp32 atomics into agg[dst].
// ---------------------------------------------------------------------------
__global__ __launch_bounds__(256) void sage_scatter(
    const int* __restrict__ src, const int* __restrict__ dst,
    const float* __restrict__ h, float* __restrict__ agg, int nedge) {
  const int lane = threadIdx.x & 31;
  int e = blockIdx.x * (blockDim.x >> 5) + (threadIdx.x >> 5);
  const int stride = gridDim.x * (blockDim.x >> 5);
  for (; e < nedge; e += stride) {
    const int s = src[e];
    const int d = dst[e];
    const float4 v = *(const float4*)(h + (size_t)s * C_DIM + lane * 4);
    float* ap = agg + (size_t)d * C_DIM + lane * 4;
    unsafeAtomicAdd(ap + 0, v.x);
    unsafeAtomicAdd(ap + 1, v.y);
    unsafeAtomicAdd(ap + 2, v.z);
    unsafeAtomicAdd(ap + 3, v.w);
  }
}

// ---------------------------------------------------------------------------
// K4: out = LayerNorm(agg @ W_l^T + b_l + x @ W_r^T).
// Two interleaved WMMA chains share one accumulator, then tile -> padded LDS,
// then 16-threads-per-row LN with shfl_xor reductions (wave32-safe masks).
// ---------------------------------------------------------------------------
__global__ __launch_bounds__(256) void sage_out_ln(
    const float* __restrict__ agg, const float* __restrict__ x,
    const float* __restrict__ Wl, const float* __restrict__ bl,
    const float* __restrict__ Wr, const float* __restrict__ gamma,
    const float* __restrict__ beta, float* __restrict__ out, int N) {
  __shared__ float st[16 * LDS_STRIDE];

  const int wave  = threadIdx.x >> 5;
  const int lane  = threadIdx.x & 31;
  const int l16   = lane & 15;
  const int khalf = (lane >> 4) * 2;
  const int row0  = blockIdx.x * 16;
  const int col   = wave * 16 + l16;

  const float* arow_a = agg + (size_t)(row0 + l16) * C_DIM;
  const float* arow_x = x   + (size_t)(row0 + l16) * C_DIM;
  const float* b_wl   = Wl  + (size_t)col * C_DIM;
  const float* b_wr   = Wr  + (size_t)col * C_DIM;

  v8f acc = {};
#pragma unroll
  for (int k0 = 0; k0 < C_DIM; k0 += 4) {
    v2f a1 = *(const v2f*)(arow_a + k0 + khalf);
    v2f b1 = *(const v2f*)(b_wl   + k0 + khalf);
    acc = wmma_f32_k4(a1, b1, acc);
    v2f a2 = *(const v2f*)(arow_x + k0 + khalf);
    v2f b2 = *(const v2f*)(b_wr   + k0 + khalf);
    acc = wmma_f32_k4(a2, b2, acc);
  }

  const float bias = bl[col];
  const int rofs = (lane >> 4) * 8;
#pragma unroll
  for (int i = 0; i < 8; ++i)
    st[(i + rofs) * LDS_STRIDE + col] = acc[i] + bias;
  __syncthreads();

  // LayerNorm: row r handled by 16 threads (t = 0..15), 8 channels each.
  const int r = threadIdx.x >> 4;
  const int t = threadIdx.x & 15;
  float vals[8];
  float s = 0.f, s2 = 0.f;
#pragma unroll
  for (int i = 0; i < 8; ++i) {
    float v = st[r * LDS_STRIDE + t + 16 * i];
    vals[i] = v;
    s += v;
    s2 += v * v;
  }
  // masks 8,4,2,1 stay within each 16-lane half of a wave32
#pragma unroll
  for (int m = 8; m >= 1; m >>= 1) {
    s  += __shfl_xor(s, m, 32);
    s2 += __shfl_xor(s2, m, 32);
  }
  const float mu   = s * (1.f / 128.f);
  const float var  = s2 * (1.f / 128.f) - mu * mu;
  const float rstd = rsqrtf(var + LN_EPS);

  float* op = out + (size_t)(row0 + r) * C_DIM;
#pragma unroll
  for (int i = 0; i < 8; ++i) {
    const int c = t + 16 * i;
    op[c] = (vals[i] - mu) * rstd * gamma[c] + beta[c];
  }
}

// ---------------------------------------------------------------------------
extern "C" void kernel_launch(void* const* d_in, const int* in_sizes, int n_in,
                              void* d_out, int out_size, void* d_ws,
                              size_t ws_size, hipStream_t stream) {
  const float* x     = (const float*)d_in[0];
  const int*   eidx  = (const int*)d_in[1];   // [2, E]
  const float* Wp    = (const float*)d_in[2];
  const float* bp    = (const float*)d_in[3];
  const float* Wl    = (const float*)d_in[4];
  const float* bl    = (const float*)d_in[5];
  const float* Wr    = (const float*)d_in[6];
  const float* gamma = (const float*)d_in[7];
  const float* beta  = (const float*)d_in[8];
  float* outp = (float*)d_out;

  const int N  = in_sizes[0] / C_DIM;  // 100000 (multiple of 16)
  const int E  = in_sizes[1] / 2;      // 1600000

  float* h   = (float*)d_ws;                   // [N, 128]
  float* agg = h + (size_t)N * C_DIM;          // [N, 128]

  const int ntiles = (N + 15) / 16;

  sage_proj_relu<<<ntiles, 256, 0, stream>>>(x, Wp, bp, h, N);

  const long n4 = (long)N * C_DIM / 4;
  sage_zero<<<2048, 256, 0, stream>>>((float4*)agg, n4);

  sage_scatter<<<8192, 256, 0, stream>>>(eidx, eidx + E, h, agg, E);

  sage_out_ln<<<ntiles, 256, 0, stream>>>(agg, x, Wl, bl, Wr, gamma, beta,
                                          outp, N);
}